// Adapter_55533927138035
// MI455X (gfx1250) — compile-verified
//
#include <hip/hip_runtime.h>

typedef float v2f __attribute__((ext_vector_type(2)));
typedef float v8f __attribute__((ext_vector_type(8)));

#define BT   128
#define SEQ  197
#define LL   196     // tokens without CLS
#define CIN  768
#define CA   384
#define TT   16
#define BB   8
#define HH   14
#define WW   14

// ---------------------------------------------------------------------------
// fp32 WMMA: D = A(16x4) * B(4x16) + C  (CDNA5 V_WMMA_F32_16X16X4_F32)
// ---------------------------------------------------------------------------
static __device__ __forceinline__ v8f wmma4(v2f a, v2f b, v8f c) {
  return __builtin_amdgcn_wmma_f32_16x16x4_f32(
      /*neg_a=*/false, a, /*neg_b=*/false, b,
      /*c_mod=*/(short)0, c, /*reuse_a=*/false, /*reuse_b=*/false);
}

// ---------------------------------------------------------------------------
// Kernel 1: fc1   h0[b,a,t,l] = sum_c x[bt,1+l,c] * fc1_w[a,c] + fc1_b[a]
// M = BT*LL = 25088, N = CA = 384, K = CIN = 768.
// One wave -> 32(M) x 64(N) tile (2x4 WMMA accumulators, 8 WMMA / K-step).
// Epilogue scatters into conv layout [B][CA][T][H*W].
// ---------------------------------------------------------------------------
__global__ __launch_bounds__(256) void fc1_wmma_kernel(
    const float* __restrict__ x, const float* __restrict__ w,
    const float* __restrict__ bias, float* __restrict__ h0)
{
  const int lane = threadIdx.x & 31;
  const int lr   = lane & 15;        // row/col within half
  const int half = lane >> 4;        // lane-half selects K offset 0 / 2
  const int gw   = blockIdx.x * 8 + (threadIdx.x >> 5);

  const int tiles_n = CA / 64;       // 6
  const int tm = gw / tiles_n;
  const int tn = gw - tm * tiles_n;
  const int m0 = tm * 32;
  const int n0 = tn * 64;

  // A row base pointers (per-lane): skip CLS token
  const int r0 = m0 + lr;
  const int r1 = r0 + 16;
  const float* arow0 = x + ((size_t)(r0 / LL) * SEQ + 1 + (r0 % LL)) * CIN;
  const float* arow1 = x + ((size_t)(r1 / LL) * SEQ + 1 + (r1 % LL)) * CIN;
  // B column base pointers: B[k][n] = fc1_w[n][k], row-major [CA][CIN]
  const float* bc0 = w + (size_t)(n0 + lr) * CIN;
  const float* bc1 = bc0 + (size_t)16 * CIN;
  const float* bc2 = bc0 + (size_t)32 * CIN;
  const float* bc3 = bc0 + (size_t)48 * CIN;

  v8f acc[2][4];
#pragma unroll
  for (int i = 0; i < 2; ++i)
#pragma unroll
    for (int j = 0; j < 4; ++j) acc[i][j] = (v8f){};

  int kb = half * 2;                 // ISA frag layout: lane halves hold K, K+2
#pragma unroll 2
  for (int k0 = 0; k0 < CIN; k0 += 4, kb += 4) {
    v2f a0 = *(const v2f*)(arow0 + kb);
    v2f a1 = *(const v2f*)(arow1 + kb);
    v2f b0 = *(const v2f*)(bc0 + kb);
    v2f b1 = *(const v2f*)(bc1 + kb);
    v2f b2 = *(const v2f*)(bc2 + kb);
    v2f b3 = *(const v2f*)(bc3 + kb);
    acc[0][0] = wmma4(a0, b0, acc[0][0]);
    acc[0][1] = wmma4(a0, b1, acc[0][1]);
    acc[0][2] = wmma4(a0, b2, acc[0][2]);
    acc[0][3] = wmma4(a0, b3, acc[0][3]);
    acc[1][0] = wmma4(a1, b0, acc[1][0]);
    acc[1][1] = wmma4(a1, b1, acc[1][1]);
    acc[1][2] = wmma4(a1, b2, acc[1][2]);
    acc[1][3] = wmma4(a1, b3, acc[1][3]);
  }

  // Epilogue: C/D layout -> VGPR r holds M = r + 8*half, N = lane&15.
#pragma unroll
  for (int i = 0; i < 2; ++i) {
#pragma unroll
    for (int j = 0; j < 4; ++j) {
      const int nc = n0 + j * 16 + lr;
      const float bv = bias[nc];
      const int mb = m0 + i * 16 + half * 8;
#pragma unroll
      for (int r = 0; r < 8; ++r) {
        const int M  = mb + r;
        const int bt = M / LL;
        const int l  = M - bt * LL;
        const int b  = bt >> 4;
        const int t  = bt & 15;
        h0[((size_t)((b * CA + nc) * TT + t)) * LL + l] = acc[i][j][r] + bv;
      }
    }
  }
}

// ---------------------------------------------------------------------------
// Kernel 2: fused depthwise convs (3x1x1 + 1x3x3 + 3x3x3)/3 + identity
// h1 = h0 + (c1(h0)+c2(h0)+c3(h0))/3 ;  layout [B][CA][T][H][W]
// ---------------------------------------------------------------------------
__global__ __launch_bounds__(256) void conv_fuse_kernel(
    const float* __restrict__ h0, float* __restrict__ h1,
    const float* __restrict__ w1, const float* __restrict__ b1,
    const float* __restrict__ w2, const float* __restrict__ b2,
    const float* __restrict__ w3, const float* __restrict__ b3)
{
  const int i = blockIdx.x * blockDim.x + threadIdx.x;
  int r = i;
  const int w = r % WW; r /= WW;
  const int h = r % HH; r /= HH;
  const int t = r % TT; r /= TT;
  const int a = r % CA;
  const int b = r / CA;

  const float* p = h0 + (size_t)(b * CA + a) * TT * LL;
  const float center = p[t * LL + h * WW + w];

  // conv1: 3x1x1 (temporal)
  float s1 = b1[a];
#pragma unroll
  for (int dt = -1; dt <= 1; ++dt) {
    const int tt = t + dt;
    if ((unsigned)tt < TT) s1 += w1[a * 3 + (dt + 1)] * p[tt * LL + h * WW + w];
  }

  // conv2: 1x3x3 (spatial)
  float s2 = b2[a];
#pragma unroll
  for (int dh = -1; dh <= 1; ++dh) {
#pragma unroll
    for (int dw = -1; dw <= 1; ++dw) {
      const int h2 = h + dh, w2i = w + dw;
      if ((unsigned)h2 < HH && (unsigned)w2i < WW)
        s2 += w2[a * 9 + (dh + 1) * 3 + (dw + 1)] * p[t * LL + h2 * WW + w2i];
    }
  }

  // conv3: 3x3x3
  float s3 = b3[a];
#pragma unroll
  for (int dt = -1; dt <= 1; ++dt) {
    const int tt = t + dt;
    if ((unsigned)tt >= TT) continue;
#pragma unroll
    for (int dh = -1; dh <= 1; ++dh) {
#pragma unroll
      for (int dw = -1; dw <= 1; ++dw) {
        const int h2 = h + dh, w2i = w + dw;
        if ((unsigned)h2 < HH && (unsigned)w2i < WW)
          s3 += w3[a * 27 + ((dt + 1) * 3 + (dh + 1)) * 3 + (dw + 1)] *
                p[tt * LL + h2 * WW + w2i];
      }
    }
  }

  h1[i] = center + (s1 + s2 + s3) / 3.0f;
}

// ---------------------------------------------------------------------------
// Kernel 3: proj dwconv (3x1x1), h2 = h1 + proj(h1); scatter to token layout
// [BT*LL][CA] so fc2 reads contiguous K.
// ---------------------------------------------------------------------------
__global__ __launch_bounds__(256) void proj_kernel(
    const float* __restrict__ h1, float* __restrict__ h2tok,
    const float* __restrict__ pw, const float* __restrict__ pb)
{
  const int i = blockIdx.x * blockDim.x + threadIdx.x;
  int r = i;
  const int a = r % CA; r /= CA;
  const int w = r % WW; r /= WW;
  const int h = r % HH; r /= HH;
  const int t = r % TT;
  const int b = r / TT;

  const float* p = h1 + (size_t)(b * CA + a) * TT * LL;
  float s = pb[a];
#pragma unroll
  for (int dt = -1; dt <= 1; ++dt) {
    const int tt = t + dt;
    if ((unsigned)tt < TT) s += pw[a * 3 + (dt + 1)] * p[tt * LL + h * WW + w];
  }
  const float val = p[t * LL + h * WW + w] + s;

  h2tok[((size_t)(b * TT + t) * LL + h * WW + w) * CA + a] = val;
}

// ---------------------------------------------------------------------------
// Kernel 4: fc2 + residual   out[bt,1+l,c] = x[bt,1+l,c] + h2tok @ fc2_w^T + fc2_b
// M = 25088, N = CIN = 768, K = CA = 384.  32x64 tile per wave.
// ---------------------------------------------------------------------------
__global__ __launch_bounds__(256) void fc2_wmma_kernel(
    const float* __restrict__ h2tok, const float* __restrict__ w,
    const float* __restrict__ bias, const float* __restrict__ x,
    float* __restrict__ out)
{
  const int lane = threadIdx.x & 31;
  const int lr   = lane & 15;
  const int half = lane >> 4;
  const int gw   = blockIdx.x * 8 + (threadIdx.x >> 5);

  const int tiles_n = CIN / 64;      // 12
  const int tm = gw / tiles_n;
  const int tn = gw - tm * tiles_n;
  const int m0 = tm * 32;
  const int n0 = tn * 64;

  const float* arow0 = h2tok + (size_t)(m0 + lr) * CA;
  const float* arow1 = arow0 + (size_t)16 * CA;
  const float* bc0 = w + (size_t)(n0 + lr) * CA;
  const float* bc1 = bc0 + (size_t)16 * CA;
  const float* bc2 = bc0 + (size_t)32 * CA;
  const float* bc3 = bc0 + (size_t)48 * CA;

  v8f acc[2][4];
#pragma unroll
  for (int i = 0; i < 2; ++i)
#pragma unroll
    for (int j = 0; j < 4; ++j) acc[i][j] = (v8f){};

  int kb = half * 2;
#pragma unroll 2
  for (int k0 = 0; k0 < CA; k0 += 4, kb += 4) {
    v2f a0 = *(const v2f*)(arow0 + kb);
    v2f a1 = *(const v2f*)(arow1 + kb);
    v2f b0 = *(const v2f*)(bc0 + kb);
    v2f b1 = *(const v2f*)(bc1 + kb);
    v2f b2 = *(const v2f*)(bc2 + kb);
    v2f b3 = *(const v2f*)(bc3 + kb);
    acc[0][0] = wmma4(a0, b0, acc[0][0]);
    acc[0][1] = wmma4(a0, b1, acc[0][1]);
    acc[0][2] = wmma4(a0, b2, acc[0][2]);
    acc[0][3] = wmma4(a0, b3, acc[0][3]);
    acc[1][0] = wmma4(a1, b0, acc[1][0]);
    acc[1][1] = wmma4(a1, b1, acc[1][1]);
    acc[1][2] = wmma4(a1, b2, acc[1][2]);
    acc[1][3] = wmma4(a1, b3, acc[1][3]);
  }

#pragma unroll
  for (int i = 0; i < 2; ++i) {
#pragma unroll
    for (int j = 0; j < 4; ++j) {
      const int nc = n0 + j * 16 + lr;
      const float bv = bias[nc];
      const int mb = m0 + i * 16 + half * 8;
#pragma unroll
      for (int r = 0; r < 8; ++r) {
        const int M  = mb + r;
        const int bt = M / LL;
        const int l  = M - bt * LL;
        const size_t oidx = ((size_t)bt * SEQ + 1 + l) * CIN + nc;
        out[oidx] = x[oidx] + acc[i][j][r] + bv;
      }
    }
  }
}

// ---------------------------------------------------------------------------
// Kernel 5: copy CLS rows through unchanged
// ---------------------------------------------------------------------------
__global__ __launch_bounds__(256) void cls_copy_kernel(
    const float* __restrict__ x, float* __restrict__ out)
{
  const int i  = blockIdx.x * blockDim.x + threadIdx.x;   // 0 .. BT*CIN-1
  const int bt = i / CIN;
  const int c  = i - bt * CIN;
  const size_t idx = (size_t)bt * SEQ * CIN + c;
  out[idx] = x[idx];
}

// ---------------------------------------------------------------------------
extern "C" void kernel_launch(void* const* d_in, const int* in_sizes, int n_in,
                              void* d_out, int out_size, void* d_ws, size_t ws_size,
                              hipStream_t stream) {
  (void)in_sizes; (void)n_in; (void)out_size; (void)ws_size;

  const float* x       = (const float*)d_in[0];
  const float* fc1_w   = (const float*)d_in[1];
  const float* fc1_b   = (const float*)d_in[2];
  const float* conv1_w = (const float*)d_in[3];
  const float* conv1_b = (const float*)d_in[4];
  const float* conv2_w = (const float*)d_in[5];
  const float* conv2_b = (const float*)d_in[6];
  const float* conv3_w = (const float*)d_in[7];
  const float* conv3_b = (const float*)d_in[8];
  const float* proj_w  = (const float*)d_in[9];
  const float* proj_b  = (const float*)d_in[10];
  const float* fc2_w   = (const float*)d_in[11];
  const float* fc2_b   = (const float*)d_in[12];
  float* outp = (float*)d_out;

  const size_t elems = (size_t)BB * CA * TT * LL;   // 9,633,792
  float* buf0 = (float*)d_ws;          // h0, later h2 (token layout)
  float* buf1 = buf0 + elems;          // h1

  // fc1: (25088/32)*(384/64) = 4704 waves -> 588 blocks of 8 waves
  fc1_wmma_kernel<<<588, 256, 0, stream>>>(x, fc1_w, fc1_b, buf0);

  // fused depthwise convs: 9,633,792 threads
  conv_fuse_kernel<<<37632, 256, 0, stream>>>(buf0, buf1,
      conv1_w, conv1_b, conv2_w, conv2_b, conv3_w, conv3_b);

  // proj conv + scatter to token layout (h2 overwrites buf0)
  proj_kernel<<<37632, 256, 0, stream>>>(buf1, buf0, proj_w, proj_b);

  // fc2 + residual: (25088/32)*(768/64) = 9408 waves -> 1176 blocks
  fc2_wmma_kernel<<<1176, 256, 0, stream>>>(buf0, fc2_w, fc2_b, x, outp);

  // CLS rows pass-through: 128*768 elements
  cls_copy_kernel<<<384, 256, 0, stream>>>(x, outp);
}